// MPNN_27891517620547
// MI455X (gfx1250) — compile-verified
//
#include <hip/hip_runtime.h>
#include <hip/hip_bf16.h>
#include <math.h>

// ---------------------------------------------------------------------------
// MPNN (NNConv x2 + GATv2/GRU readout) for MI455X / gfx1250.
// All dense math runs on V_WMMA_F32_16X16X4_F32 (fp32 matrix pipe, exact
// vs. the fp32 reference). The edge-conditioned message pass is fused into a
// single bilinear GEMM  A'[E,4096] @ W2[4096,64]  with A' formed in-register
// from LDS tiles of t=relu(ea@W1+b1) and gathered h[src] -- avoiding the
// 1 GB 'ew' intermediate. Each wave owns 128 edges (8 M-tiles) so every
// W2 B-operand fetch is amortized across 8 WMMAs, halving L2 traffic vs. a
// 4-M-tile scheme. W2 rows are prefetched 8 k-steps ahead.
//
// NOTE: edge_msg_kernel is defined FIRST so the compile-loop asm snippet
// shows its prologue (VGPR count / any scratch spill / prefetch lowering).
// ---------------------------------------------------------------------------

typedef float v2f __attribute__((ext_vector_type(2)));
typedef float v8f __attribute__((ext_vector_type(8)));

#define N_  32768
#define E_  65536
#define B_  1024
#define DH_ 64
#define NEG_ 0.01f

#define MT_  8            // M-tiles (16 edges each) per wave in edge kernel
#define EPW_ (MT_ * 16)   // 128 edges per wave

__device__ inline v8f splat8(float x) { v8f r = {x,x,x,x,x,x,x,x}; return r; }

// D = A(16x4) * B(4x16) + C(16x16), fp32. Assumed lane layout (ISA 7.12.2):
//   A: lane<16 -> {K0,K1}, lane>=16 -> {K2,K3}, row = lane&15
//   B: mirrored on K, col = lane&15
//   C/D: vgpr v -> row v + 8*(lane>>4), col = lane&15
__device__ inline v8f wmma4(v2f a, v2f b, v8f c) {
  return __builtin_amdgcn_wmma_f32_16x16x4_f32(false, a, false, b,
                                               (short)0, c, false, false);
}

__device__ inline float lre(float v) { return v > 0.0f ? v : NEG_ * v; }

// monotonic float<->uint encoding for atomicMax on floats
__device__ inline unsigned fenc(float x) {
  unsigned u = __float_as_uint(x);
  return (u & 0x80000000u) ? ~u : (u | 0x80000000u);
}
__device__ inline float fdec(unsigned u) {
  u = (u & 0x80000000u) ? (u & 0x7FFFFFFFu) : ~u;
  return __uint_as_float(u);
}

// ---------------------------------------------------------------------------
// Fused NNConv message kernel. One wave per block owns 128 edges (8 M-tiles):
//   t  = relu(ea @ W1 + b1)          (WMMA, stored to LDS)
//   hs = h[src]                      (gathered into LDS)
//   msg= hs @ b2v  +  A' @ W2v       (WMMA; A'[e,k*64+i] = t[e,k]*hs[e,i])
//   agg[dst] += msg                  (fp32 global atomics)
// ---------------------------------------------------------------------------
__global__ void __launch_bounds__(32)
edge_msg_kernel(const float* __restrict__ ea, const float* __restrict__ h,
                const int* __restrict__ eidx,
                const float* __restrict__ W1, const float* __restrict__ b1,
                const float* __restrict__ W2, const float* __restrict__ b2,
                float* __restrict__ agg)
{
  __shared__ float sT[EPW_][DH_];    // t tile (128 x 64 fp32, 32 KB)
  __shared__ float sH[EPW_][DH_];    // gathered h[src]   (32 KB)

  const int lane  = threadIdx.x & 31;
  const int ebase = blockIdx.x * EPW_;
  const int row16 = lane & 15;
  const int hi    = lane >> 4;

  // gather h[src] rows -> LDS (float4 per lane-iteration)
  for (int idx = lane; idx < EPW_ * 16; idx += 32) {
    const int r = idx >> 4, q = idx & 15;
    const int s = eidx[ebase + r];
    float4 v = *(const float4*)(h + (size_t)s * DH_ + q * 4);
    *(float4*)(&sH[r][q * 4]) = v;
  }

  // t = relu(ea @ W1 + b1), 8 M-tiles, K=16
#pragma unroll
  for (int mt = 0; mt < MT_; ++mt) {
    const float* Arow = ea + (size_t)(ebase + mt * 16 + row16) * 16;
    v8f tacc[4];
#pragma unroll
    for (int nt = 0; nt < 4; ++nt) tacc[nt] = splat8(b1[nt * 16 + row16]);
#pragma unroll
    for (int ks = 0; ks < 4; ++ks) {
      const int k0 = ks * 4 + 2 * hi;
      v2f a = { Arow[k0], Arow[k0 + 1] };
      const float* Wr = W1 + (size_t)k0 * DH_;
#pragma unroll
      for (int nt = 0; nt < 4; ++nt) {
        v2f b = { Wr[nt * 16 + row16], Wr[DH_ + nt * 16 + row16] };
        tacc[nt] = wmma4(a, b, tacc[nt]);
      }
    }
#pragma unroll
    for (int nt = 0; nt < 4; ++nt)
#pragma unroll
      for (int v = 0; v < 8; ++v)
        sT[mt * 16 + v + 8 * hi][nt * 16 + row16] = fmaxf(tacc[nt][v], 0.0f);
  }

  v8f acc[MT_][4];
#pragma unroll
  for (int mt = 0; mt < MT_; ++mt)
#pragma unroll
    for (int nt = 0; nt < 4; ++nt) acc[mt][nt] = splat8(0.0f);

  // bias term: msg += hs @ reshape(b2,[64,64]) ; K = 64
  for (int ks = 0; ks < 16; ++ks) {
    const int i0 = ks * 4 + 2 * hi;
    v2f a[MT_];
#pragma unroll
    for (int mt = 0; mt < MT_; ++mt) {
      const int rr = mt * 16 + row16;
      a[mt].x = sH[rr][i0];
      a[mt].y = sH[rr][i0 + 1];
    }
    const float* Br = b2 + (size_t)i0 * DH_;
#pragma unroll
    for (int nt = 0; nt < 4; ++nt) {
      v2f b = { Br[nt * 16 + row16], Br[DH_ + nt * 16 + row16] };
#pragma unroll
      for (int mt = 0; mt < MT_; ++mt) acc[mt][nt] = wmma4(a[mt], b, acc[mt][nt]);
    }
  }

  // main bilinear term: K' = 4096, A'[e, k*64+i] = t[e,k] * hs[e,i]
  float tc[MT_];
#pragma unroll
  for (int mt = 0; mt < MT_; ++mt) tc[mt] = 0.0f;

  for (int ks = 0; ks < 1024; ++ks) {
    const int r4 = ks * 4;
    if ((ks & 15) == 0) {
      const int k = r4 >> 6;
#pragma unroll
      for (int mt = 0; mt < MT_; ++mt) tc[mt] = sT[mt * 16 + row16][k];
    }
    // prefetch W2 rows 8 k-steps (32 rows) ahead (global_prefetch_b8);
    // guarded so speculation never walks past the 1 MB W2 buffer.
    if ((ks & 7) == 0 && r4 < 4032) {
      const float* pf = W2 + (size_t)(r4 + 32 + lane) * DH_;
      __builtin_prefetch(pf, 0, 3);
      __builtin_prefetch(pf + 32, 0, 3);
    }
    const int i0 = (r4 & 63) + 2 * hi;
    v2f a[MT_];
#pragma unroll
    for (int mt = 0; mt < MT_; ++mt) {
      const int rr = mt * 16 + row16;
      a[mt].x = tc[mt] * sH[rr][i0];
      a[mt].y = tc[mt] * sH[rr][i0 + 1];
    }
    const float* Wr = W2 + (size_t)(r4 + 2 * hi) * DH_;   // W2 as [4096,64]
#pragma unroll
    for (int nt = 0; nt < 4; ++nt) {
      v2f b = { Wr[nt * 16 + row16], Wr[DH_ + nt * 16 + row16] };
#pragma unroll
      for (int mt = 0; mt < MT_; ++mt) acc[mt][nt] = wmma4(a[mt], b, acc[mt][nt]);
    }
  }

  // scatter-add to agg[dst]
#pragma unroll
  for (int mt = 0; mt < MT_; ++mt) {
    int dv[8];
#pragma unroll
    for (int v = 0; v < 8; ++v)
      dv[v] = eidx[E_ + ebase + mt * 16 + v + 8 * hi];
#pragma unroll
    for (int nt = 0; nt < 4; ++nt)
#pragma unroll
      for (int v = 0; v < 8; ++v)
        atomicAdd(&agg[(size_t)dv[v] * DH_ + nt * 16 + row16], acc[mt][nt][v]);
  }
}

// ---------------------------------------------------------------------------
// Generic wave-level GEMM: D[M,64] = act(A[M,K] @ W[K,64] + bias (+ addend))
// One wave computes a 16-row tile across all 64 output columns.
// act: 0 = none, 1 = leaky_relu(0.01), 2 = relu
// ---------------------------------------------------------------------------
__global__ void __launch_bounds__(128)
fc_wmma_kernel(const float* __restrict__ A, int lda,
               const float* __restrict__ W,
               const float* __restrict__ bias,
               const float* __restrict__ addend,
               float* __restrict__ D,
               int M, int K, int act)
{
  const int lane = threadIdx.x & 31;
  const int wid  = threadIdx.x >> 5;
  const int tile = blockIdx.x * 4 + wid;
  if (tile * 16 >= M) return;          // wave-uniform: EXEC stays all-ones
  const int col = lane & 15;
  const int hi  = lane >> 4;

  const float* Arow = A + (size_t)(tile * 16 + col) * lda;

  v8f acc[4];
#pragma unroll
  for (int nt = 0; nt < 4; ++nt)
    acc[nt] = splat8(bias ? bias[nt * 16 + col] : 0.0f);

  const int nks = K >> 2;
  for (int ks = 0; ks < nks; ++ks) {
    const int k0 = ks * 4 + 2 * hi;
    v2f a = { Arow[k0], Arow[k0 + 1] };
    const float* Wr = W + (size_t)k0 * DH_;
#pragma unroll
    for (int nt = 0; nt < 4; ++nt) {
      v2f b = { Wr[nt * 16 + col], Wr[DH_ + nt * 16 + col] };
      acc[nt] = wmma4(a, b, acc[nt]);
    }
  }

#pragma unroll
  for (int nt = 0; nt < 4; ++nt) {
#pragma unroll
    for (int v = 0; v < 8; ++v) {
      const int m = tile * 16 + v + 8 * hi;
      const int c = nt * 16 + col;
      float val = acc[nt][v];
      if (addend) val += addend[(size_t)m * DH_ + c];
      if (act == 1)      val = val > 0.0f ? val : NEG_ * val;
      else if (act == 2) val = fmaxf(val, 0.0f);
      D[(size_t)m * DH_ + c] = val;
    }
  }
}

// ---------------------------------------------------------------------------
// Small elementwise / segment kernels (readout tail, <1% of FLOPs)
// ---------------------------------------------------------------------------
__global__ void normalize_kernel(float* __restrict__ h) {
  const int n = blockIdx.x * blockDim.x + threadIdx.x;
  if (n >= N_) return;
  float* r = h + (size_t)n * DH_;
  float ss = 0.f;
  for (int o = 0; o < DH_; ++o) { float v = r[o]; ss += v * v; }
  const float s = 1.0f / fmaxf(sqrtf(ss), 1e-12f);
  for (int o = 0; o < DH_; ++o) r[o] *= s;
}

__global__ void pool_sum_kernel(const float* __restrict__ h,
                                const int* __restrict__ batch,
                                float* __restrict__ sum) {
  const int i = blockIdx.x * blockDim.x + threadIdx.x;
  if (i >= N_ * DH_) return;
  const int n = i >> 6, o = i & 63;
  atomicAdd(&sum[(size_t)batch[n] * DH_ + o], h[i]);
}

__global__ void relu_copy_kernel(const float* __restrict__ a,
                                 float* __restrict__ d, int count) {
  const int i = blockIdx.x * blockDim.x + threadIdx.x;
  if (i < count) d[i] = fmaxf(a[i], 0.0f);
}

__global__ void init_segmax_kernel(unsigned* __restrict__ sm) {
  const int b = blockIdx.x * blockDim.x + threadIdx.x;
  if (b < B_) sm[b] = fenc(-3.0e38f);
}

__global__ void alpha_kernel(const float* __restrict__ xs,
                             const float* __restrict__ xd,
                             const int* __restrict__ batch,
                             const float* __restrict__ att,
                             float* __restrict__ alpha,
                             unsigned* __restrict__ segmax) {
  const int n = blockIdx.x * blockDim.x + threadIdx.x;
  if (n >= N_) return;
  const int b = batch[n];
  const float* xr = xs + (size_t)n * DH_;
  const float* dr = xd + (size_t)b * DH_;
  float s = 0.f;
  for (int o = 0; o < DH_; ++o) s += lre(xr[o] + dr[o]) * att[o];
  alpha[n] = s;
  atomicMax(&segmax[b], fenc(s));
}

__global__ void w_kernel(const float* __restrict__ alpha,
                         const unsigned* __restrict__ segmax,
                         const int* __restrict__ batch,
                         float* __restrict__ wv, float* __restrict__ den) {
  const int n = blockIdx.x * blockDim.x + threadIdx.x;
  if (n >= N_) return;
  const int b = batch[n];
  const float v = expf(alpha[n] - fdec(segmax[b]));
  wv[n] = v;
  atomicAdd(&den[b], v);
}

__global__ void hg_accum_kernel(const float* __restrict__ xs,
                                const float* __restrict__ wv,
                                const float* __restrict__ den,
                                const int* __restrict__ batch,
                                float* __restrict__ hgs) {
  const int i = blockIdx.x * blockDim.x + threadIdx.x;
  if (i >= N_ * DH_) return;
  const int n = i >> 6, o = i & 63, b = batch[n];
  const float a = wv[n] / (den[b] + 1e-16f);
  atomicAdd(&hgs[(size_t)b * DH_ + o], xs[i] * a);
}

__global__ void hg_elu_kernel(const float* __restrict__ hgs,
                              const float* __restrict__ bias,
                              float* __restrict__ hg) {
  const int i = blockIdx.x * blockDim.x + threadIdx.x;
  if (i >= B_ * DH_) return;
  const float v = hgs[i] + bias[i & 63];
  hg[i] = v > 0.0f ? v : (expf(v) - 1.0f);
}

__global__ void gru_kernel(const float* __restrict__ hg,
                           const float* __restrict__ oldo,
                           const float* __restrict__ Wih,
                           const float* __restrict__ Whh,
                           const float* __restrict__ bih,
                           const float* __restrict__ bhh,
                           float* __restrict__ newo) {
  const int i = blockIdx.x * blockDim.x + threadIdx.x;
  if (i >= B_ * DH_) return;
  const int b = i >> 6, j = i & 63;
  const float* hrow = hg + (size_t)b * DH_;
  const float* orow = oldo + (size_t)b * DH_;
  float ir = bih[j], iz = bih[64 + j], inn = bih[128 + j];
  float hr = bhh[j], hz = bhh[64 + j], hn  = bhh[128 + j];
  const float* w0 = Wih + (size_t)j * DH_;
  const float* w1 = Wih + (size_t)(64 + j) * DH_;
  const float* w2 = Wih + (size_t)(128 + j) * DH_;
  const float* v0 = Whh + (size_t)j * DH_;
  const float* v1 = Whh + (size_t)(64 + j) * DH_;
  const float* v2 = Whh + (size_t)(128 + j) * DH_;
  for (int k = 0; k < DH_; ++k) {
    const float hv = hrow[k], ov = orow[k];
    ir += hv * w0[k]; iz += hv * w1[k]; inn += hv * w2[k];
    hr += ov * v0[k]; hz += ov * v1[k]; hn  += ov * v2[k];
  }
  const float r  = 1.0f / (1.0f + expf(-(ir + hr)));
  const float z  = 1.0f / (1.0f + expf(-(iz + hz)));
  const float nn = tanhf(inn + r * hn);
  newo[i] = fmaxf((1.0f - z) * nn + z * orow[j], 0.0f);
}

__global__ void final_kernel(const float* __restrict__ tmp,
                             const float* __restrict__ fcW,
                             const float* __restrict__ fcb,
                             float* __restrict__ out) {
  const int b = blockIdx.x * blockDim.x + threadIdx.x;
  if (b >= B_) return;
  const float* tr = tmp + (size_t)b * DH_;
  float s = fcb[0];
  for (int o = 0; o < DH_; ++o) s += tr[o] * fcW[o];
  out[b] = s;
}

// ---------------------------------------------------------------------------
static void launch_fc(const float* A, int lda, const float* W, const float* bias,
                      const float* addend, float* D, int M, int K, int act,
                      hipStream_t s) {
  const int tiles = M / 16;
  const int blocks = (tiles + 3) / 4;
  fc_wmma_kernel<<<blocks, 128, 0, s>>>(A, lda, W, bias, addend, D, M, K, act);
}

extern "C" void kernel_launch(void* const* d_in, const int* in_sizes, int n_in,
                              void* d_out, int out_size, void* d_ws, size_t ws_size,
                              hipStream_t stream) {
  const float* x       = (const float*)d_in[0];
  const float* ea      = (const float*)d_in[1];
  const int*   eidx    = (const int*)  d_in[2];
  const int*   batch   = (const int*)  d_in[3];
  const float* nfc_W   = (const float*)d_in[4];
  const float* nfc_b   = (const float*)d_in[5];
  const float* e1W1    = (const float*)d_in[6];
  const float* e1b1    = (const float*)d_in[7];
  const float* e1W2    = (const float*)d_in[8];
  const float* e1b2    = (const float*)d_in[9];
  const float* g1root  = (const float*)d_in[10];
  const float* g1b     = (const float*)d_in[11];
  const float* e2W1    = (const float*)d_in[12];
  const float* e2b1    = (const float*)d_in[13];
  const float* e2W2    = (const float*)d_in[14];
  const float* e2b2    = (const float*)d_in[15];
  const float* g2root  = (const float*)d_in[16];
  const float* g2b     = (const float*)d_in[17];
  const float* gat_Wl  = (const float*)d_in[18];
  const float* gat_bl  = (const float*)d_in[19];
  const float* gat_Wr  = (const float*)d_in[20];
  const float* gat_br  = (const float*)d_in[21];
  const float* gat_att = (const float*)d_in[22];
  const float* gat_bias= (const float*)d_in[23];
  const float* gru_Wih = (const float*)d_in[24];
  const float* gru_Whh = (const float*)d_in[25];
  const float* gru_bih = (const float*)d_in[26];
  const float* gru_bhh = (const float*)d_in[27];
  const float* fco_W   = (const float*)d_in[28];
  const float* fco_b   = (const float*)d_in[29];
  const float* fc_W    = (const float*)d_in[30];
  const float* fc_b    = (const float*)d_in[31];

  float* ws    = (float*)d_ws;
  float* bufA  = ws;                       // N*64  node features
  float* bufB  = bufA + (size_t)N_ * DH_;  // N*64  agg / xs
  float* bufC  = bufB + (size_t)N_ * DH_;  // N*64  node features (ping-pong)
  float* outA  = bufC + (size_t)N_ * DH_;  // B*64
  float* outB  = outA + (size_t)B_ * DH_;  // B*64
  float* xd    = outB + (size_t)B_ * DH_;  // B*64
  float* hgs   = xd   + (size_t)B_ * DH_;  // B*64
  float* hg    = hgs  + (size_t)B_ * DH_;  // B*64
  float* tmp   = hg   + (size_t)B_ * DH_;  // B*64
  float* alpha = tmp  + (size_t)B_ * DH_;  // N
  float* wv    = alpha + (size_t)N_;       // N
  unsigned* segmax = (unsigned*)(wv + (size_t)N_);   // B
  float* den   = (float*)(segmax + B_);              // B

  const size_t nodeBytes  = (size_t)N_ * DH_ * sizeof(float);
  const size_t graphBytes = (size_t)B_ * DH_ * sizeof(float);

  // h = lre(x @ nfc_W + nfc_b)
  launch_fc(x, 32, nfc_W, nfc_b, nullptr, bufA, N_, 32, 1, stream);

  // --- NNConv layer 1 ---
  hipMemsetAsync(bufB, 0, nodeBytes, stream);
  edge_msg_kernel<<<E_ / EPW_, 32, 0, stream>>>(ea, bufA, eidx, e1W1, e1b1,
                                                e1W2, e1b2, bufB);
  launch_fc(bufA, DH_, g1root, g1b, bufB, bufC, N_, DH_, 1, stream);

  // --- NNConv layer 2 ---
  hipMemsetAsync(bufB, 0, nodeBytes, stream);
  edge_msg_kernel<<<E_ / EPW_, 32, 0, stream>>>(ea, bufC, eidx, e2W1, e2b1,
                                                e2W2, e2b2, bufB);
  launch_fc(bufC, DH_, g2root, g2b, bufB, bufA, N_, DH_, 1, stream);

  // h = F.normalize(h)
  normalize_kernel<<<N_ / 256, 256, 0, stream>>>(bufA);

  // xs = h @ gat_Wl + gat_bl
  launch_fc(bufA, DH_, gat_Wl, gat_bl, nullptr, bufB, N_, DH_, 0, stream);

  // out = relu(global_add_pool(h))
  hipMemsetAsync(outB, 0, graphBytes, stream);
  pool_sum_kernel<<<(N_ * DH_) / 256, 256, 0, stream>>>(bufA, batch, outB);
  relu_copy_kernel<<<(B_ * DH_) / 256, 256, 0, stream>>>(outB, outA, B_ * DH_);

  float* cur = outA;
  float* nxt = outB;
  for (int t = 0; t < 4; ++t) {
    launch_fc(cur, DH_, gat_Wr, gat_br, nullptr, xd, B_, DH_, 0, stream);
    init_segmax_kernel<<<(B_ + 255) / 256, 256, 0, stream>>>(segmax);
    alpha_kernel<<<N_ / 256, 256, 0, stream>>>(bufB, xd, batch, gat_att,
                                               alpha, segmax);
    hipMemsetAsync(den, 0, B_ * sizeof(float), stream);
    w_kernel<<<N_ / 256, 256, 0, stream>>>(alpha, segmax, batch, wv, den);
    hipMemsetAsync(hgs, 0, graphBytes, stream);
    hg_accum_kernel<<<(N_ * DH_) / 256, 256, 0, stream>>>(bufB, wv, den,
                                                          batch, hgs);
    hg_elu_kernel<<<(B_ * DH_) / 256, 256, 0, stream>>>(hgs, gat_bias, hg);
    gru_kernel<<<(B_ * DH_) / 256, 256, 0, stream>>>(hg, cur, gru_Wih, gru_Whh,
                                                     gru_bih, gru_bhh, nxt);
    float* sw = cur; cur = nxt; nxt = sw;
  }

  // final projections
  launch_fc(cur, DH_, fco_W, fco_b, nullptr, tmp, B_, DH_, 0, stream);
  final_kernel<<<(B_ + 255) / 256, 256, 0, stream>>>(tmp, fc_W, fc_b,
                                                     (float*)d_out);
}